// GRUWeatherForecaster_38173669327477
// MI455X (gfx1250) — compile-verified
//
#include <hip/hip_runtime.h>

// GRU recurrence on CDNA5 using V_WMMA_F32_16X16X4_F32 + V_TANH_F32.
// One wave (32 threads) owns 16 batch rows and runs all 512 timesteps.
// Per step: 18 chained f32 WMMAs (max dependency depth 4), gates via native tanh.

typedef float v2f __attribute__((ext_vector_type(2)));
typedef float v8f __attribute__((ext_vector_type(8)));

namespace {
constexpr int Tn = 512;
constexpr int In = 5;
constexpr int Hn = 16;
constexpr int ROWS = 18;  // LDS row stride (floats): conflict-free strided reads, 8B aligned
}

__device__ __forceinline__ float fast_tanh(float a) {
#if __has_builtin(__builtin_amdgcn_tanhf)
    return __builtin_amdgcn_tanhf(a);           // v_tanh_f32 (gfx1250 TRANS op)
#else
    // tanh(a) = 2/(1+exp(-2a)) - 1, with single-instruction v_rcp_f32
    const float e = __expf(-2.0f * a);
    return __builtin_amdgcn_rcpf(1.0f + e) * 2.0f - 1.0f;
#endif
}

__device__ __forceinline__ float fast_sigmoid(float a) {
    // exact identity: sigmoid(x) = 0.5*tanh(x/2) + 0.5
    return fmaf(0.5f, fast_tanh(0.5f * a), 0.5f);
}

__global__ __launch_bounds__(32)
void gru_fused_kernel(const float* __restrict__ x,
                      const float* __restrict__ W_ih,   // (48,5)
                      const float* __restrict__ W_hh,   // (48,16)
                      const float* __restrict__ b_ih,   // (48)
                      const float* __restrict__ b_hh,   // (48)
                      const float* __restrict__ W_fc,   // (1,16)
                      const float* __restrict__ b_fc,   // (1)
                      float* __restrict__ out)          // (B,1)
{
    __shared__ __align__(16) float hs[16 * ROWS];

    const int lane = threadIdx.x;        // 0..31
    const int half = lane >> 4;          // 0/1: selects K sub-pair in A/B frags
    const int col  = lane & 15;          // M (for A) / N (for B,C,D)
    const long b0  = (long)blockIdx.x * 16;

    // ---- B fragments for W_hh: Bh[g][c] covers K = c*4..c*4+3 of gh = h @ W_hh^T
    v2f Bh[3][4];
    #pragma unroll
    for (int g = 0; g < 3; ++g) {
        const int row = g * 16 + col;
        #pragma unroll
        for (int c = 0; c < 4; ++c) {
            Bh[g][c].x = W_hh[row * 16 + c * 4 + 2 * half];
            Bh[g][c].y = W_hh[row * 16 + c * 4 + 2 * half + 1];
        }
    }

    // ---- B fragments for W_ih (K = 0..4 valid, padded to 8)
    const float hmask = (half == 0) ? 1.0f : 0.0f;
    v2f Bx0[3], Bx1[3];
    #pragma unroll
    for (int g = 0; g < 3; ++g) {
        const int row = g * 16 + col;
        Bx0[g].x = W_ih[row * In + 2 * half];
        Bx0[g].y = W_ih[row * In + 2 * half + 1];
        Bx1[g].x = W_ih[row * In + 4] * hmask;  // K=4 valid only in lanes 0..15
        Bx1[g].y = 0.0f;                        // K=5..7 padding
    }

    // ---- persistent broadcast bias / zero accumulator seeds (no per-step movs)
    const float biasR  = b_ih[col]      + b_hh[col];
    const float biasZ  = b_ih[16 + col] + b_hh[16 + col];
    const float biasXN = b_ih[32 + col];
    const float biasHN = b_hh[32 + col];
    v8f biasRv, biasZv, biasXNv, biasHNv, zerov;
    #pragma unroll
    for (int j = 0; j < 8; ++j) {
        biasRv[j] = biasR; biasZv[j] = biasZ;
        biasXNv[j] = biasXN; biasHNv[j] = biasHN;
        zerov[j] = 0.0f;
    }

    // per-lane x row pointer: A-matrix row M = col -> batch b0+col
    const float* __restrict__ px = x + (b0 + col) * (long)(Tn * In);

    // hidden state in D layout: hD[j] = h[m = j + 8*half][n = col]
    v8f hD;
    #pragma unroll
    for (int j = 0; j < 8; ++j) hD[j] = 0.0f;

    #pragma unroll 1
    for (int t = 0; t < Tn; ++t) {
        // publish h (D layout) to LDS so it can be re-read in A layout
        #pragma unroll
        for (int j = 0; j < 8; ++j)
            hs[(j + 8 * half) * ROWS + col] = hD[j];
        __syncthreads();  // single-wave WG: ordering fence only

        // A fragments of h: Ah[c] holds h[M=col][K = c*4 + 2*half, +1]
        v2f Ah[4];
        #pragma unroll
        for (int c = 0; c < 4; ++c) {
            Ah[c].x = hs[col * ROWS + c * 4 + 2 * half];
            Ah[c].y = hs[col * ROWS + c * 4 + 2 * half + 1];
        }

        // A fragments of x[t]: K=0..4 valid, padded to 8
        const float* __restrict__ pxt = px + t * In;
        v2f Ax0, Ax1;
        Ax0.x = pxt[2 * half];
        Ax0.y = pxt[2 * half + 1];
        Ax1.x = pxt[4] * hmask;
        Ax1.y = 0.0f;

        if (t + 8 < Tn) __builtin_prefetch(px + (t + 8) * In, 0, 1);

        // ---- six independent WMMA chains, max depth 4
        // r-gate
        v8f Crx = __builtin_amdgcn_wmma_f32_16x16x4_f32(false, Ax0, false, Bx0[0], (short)0, biasRv, false, false);
        Crx     = __builtin_amdgcn_wmma_f32_16x16x4_f32(false, Ax1, false, Bx1[0], (short)0, Crx,    false, false);
        v8f Crh = __builtin_amdgcn_wmma_f32_16x16x4_f32(false, Ah[0], false, Bh[0][0], (short)0, zerov, false, false);
        #pragma unroll
        for (int c = 1; c < 4; ++c)
            Crh = __builtin_amdgcn_wmma_f32_16x16x4_f32(false, Ah[c], false, Bh[0][c], (short)0, Crh, false, false);

        // z-gate
        v8f Czx = __builtin_amdgcn_wmma_f32_16x16x4_f32(false, Ax0, false, Bx0[1], (short)0, biasZv, false, false);
        Czx     = __builtin_amdgcn_wmma_f32_16x16x4_f32(false, Ax1, false, Bx1[1], (short)0, Czx,    false, false);
        v8f Czh = __builtin_amdgcn_wmma_f32_16x16x4_f32(false, Ah[0], false, Bh[1][0], (short)0, zerov, false, false);
        #pragma unroll
        for (int c = 1; c < 4; ++c)
            Czh = __builtin_amdgcn_wmma_f32_16x16x4_f32(false, Ah[c], false, Bh[1][c], (short)0, Czh, false, false);

        // n-gate: gx_n and gh_n kept separate (r scales only gh_n)
        v8f Cxn = __builtin_amdgcn_wmma_f32_16x16x4_f32(false, Ax0, false, Bx0[2], (short)0, biasXNv, false, false);
        Cxn     = __builtin_amdgcn_wmma_f32_16x16x4_f32(false, Ax1, false, Bx1[2], (short)0, Cxn,     false, false);
        v8f Chn = __builtin_amdgcn_wmma_f32_16x16x4_f32(false, Ah[0], false, Bh[2][0], (short)0, biasHNv, false, false);
        #pragma unroll
        for (int c = 1; c < 4; ++c)
            Chn = __builtin_amdgcn_wmma_f32_16x16x4_f32(false, Ah[c], false, Bh[2][c], (short)0, Chn, false, false);

        // ---- elementwise gates + state update, in D layout (tanh = TRANS, co-executes)
        #pragma unroll
        for (int j = 0; j < 8; ++j) {
            const float r = fast_sigmoid(Crx[j] + Crh[j]);
            const float z = fast_sigmoid(Czx[j] + Czh[j]);
            const float n = fast_tanh(Cxn[j] + r * Chn[j]);
            hD[j] = n + z * (hD[j] - n);             // (1-z)*n + z*h
        }
        __syncthreads();  // WAR fence before next iteration's LDS stores
    }

    // ---- final projection: out[b] = h_T[b,:] . W_fc + b_fc
    #pragma unroll
    for (int j = 0; j < 8; ++j)
        hs[(j + 8 * half) * ROWS + col] = hD[j];
    __syncthreads();

    if (lane < 16) {
        float acc = b_fc[0];
        #pragma unroll
        for (int n = 0; n < Hn; ++n)
            acc += hs[lane * ROWS + n] * W_fc[n];
        out[b0 + lane] = acc;
    }
}

extern "C" void kernel_launch(void* const* d_in, const int* in_sizes, int n_in,
                              void* d_out, int out_size, void* d_ws, size_t ws_size,
                              hipStream_t stream) {
    const float* x    = (const float*)d_in[0];
    const float* W_ih = (const float*)d_in[1];
    const float* W_hh = (const float*)d_in[2];
    const float* b_ih = (const float*)d_in[3];
    const float* b_hh = (const float*)d_in[4];
    const float* W_fc = (const float*)d_in[5];
    const float* b_fc = (const float*)d_in[6];
    float* out = (float*)d_out;

    const int B = in_sizes[0] / (Tn * In);   // 8192
    const int tiles = (B + 15) / 16;         // 512 waves, one per 16 batch rows

    gru_fused_kernel<<<dim3(tiles), dim3(32), 0, stream>>>(
        x, W_ih, W_hh, b_ih, b_hh, W_fc, b_fc, out);
}